// local_net_53523882443072
// MI455X (gfx1250) — compile-verified
//
#include <hip/hip_runtime.h>
#include <hip/hip_bf16.h>

// Problem constants (match reference).
#define H 2048
#define W 2048
#define HW (H * W)

typedef float v4f __attribute__((ext_vector_type(4)));

// Each thread computes 4 consecutive output pixels (i, j0..j0+3).
// cg[i,j] = c00*(v[i-1,j]-v[i,j]) + c01*(v[i,j-1]-v[i,j])
//         + c10*(v[i,j+1]-v[i,j]) + c11*(v[i+1,j]-v[i,j])
// with missing-neighbor terms zeroed at the borders (unified form of all
// boundary slices in voltage_transform; vmat_bound's dir-writes are bitwise
// no-ops in the forward pass), cXY = conduct[2i+X, 2j+Y].
//
// conduct (64 MB, streamed once) is fetched with CDNA5 async-tensor path:
// GLOBAL_LOAD_ASYNC_TO_LDS_B128 (ASYNCcnt) issued up front; the v loads and
// all difference VALU work overlap the async stream; then s_wait_asynccnt 0
// and ds_load_b128 readback from each thread's private 64B LDS slot.
//
// Outputs (concatenated): loss | cur_loss | cg | cur_active.
__global__ __launch_bounds__(256) void grid_current_kernel(
    const float* __restrict__ v,
    const float* __restrict__ conduct,
    const float* __restrict__ weight,
    float* __restrict__ out)
{
    __shared__ float cbuf[256 * 16];   // 16 KB: 64 B per thread

    const int i  = blockIdx.y;
    const int jt = blockIdx.x * blockDim.x + threadIdx.x;   // 0 .. W/4-1
    const int j0 = jt * 4;                                  // grid covers W/4 exactly

    // ---- Stage conduct via async copy to LDS (issued first, completes in
    // background while we do the v loads + stencil arithmetic). The inst
    // offset is added to BOTH the global and LDS address, so {+0,+16} covers
    // a 32B chunk pair per conduct row.
    const size_t cw  = (size_t)(2 * W);
    const float* cr0 = conduct + (size_t)(2 * i) * cw + (size_t)(2 * j0);
    const float* cr1 = cr0 + cw;
    float* myslot = cbuf + (size_t)threadIdx.x * 16;
    const unsigned l0 = (unsigned)(uintptr_t)myslot;        // LDS aperture: addr[31:0] = LDS byte offset
    const unsigned l1 = l0 + 32u;
    asm volatile(
        "global_load_async_to_lds_b128 %0, %2, off th:TH_LOAD_NT\n\t"
        "global_load_async_to_lds_b128 %0, %2, off offset:16 th:TH_LOAD_NT\n\t"
        "global_load_async_to_lds_b128 %1, %3, off th:TH_LOAD_NT\n\t"
        "global_load_async_to_lds_b128 %1, %3, off offset:16 th:TH_LOAD_NT"
        :: "v"(l0), "v"(l1), "v"(cr0), "v"(cr1) : "memory");

    // ---- v is re-read across neighboring rows -> regular-temporal (L2-resident).
    const float* vrow = v + (size_t)i * W;
    const v4f vc = *(const v4f*)(vrow + j0);
    const v4f zero4 = {0.f, 0.f, 0.f, 0.f};
    const v4f vn = (i > 0)     ? *(const v4f*)(vrow - W + j0) : zero4;
    const v4f vs = (i < H - 1) ? *(const v4f*)(vrow + W + j0) : zero4;
    const float vwesthalo = (j0 > 0)     ? vrow[j0 - 1] : 0.f;
    const float veasthalo = (j0 + 4 < W) ? vrow[j0 + 4] : 0.f;

    // weight: streamed once -> non-temporal.
    const v4f wgt =
        __builtin_nontemporal_load((const v4f*)(weight + (size_t)i * W + j0));

    const float c0 = vc.x, c1 = vc.y, c2 = vc.z, c3 = vc.w;

    // Neighbor differences with boundary zeroing (overlaps the async copy).
    const bool hasN = (i > 0), hasS = (i < H - 1);
    const float nd0 = hasN ? vn.x - c0 : 0.f;
    const float nd1 = hasN ? vn.y - c1 : 0.f;
    const float nd2 = hasN ? vn.z - c2 : 0.f;
    const float nd3 = hasN ? vn.w - c3 : 0.f;
    const float sd0 = hasS ? vs.x - c0 : 0.f;
    const float sd1 = hasS ? vs.y - c1 : 0.f;
    const float sd2 = hasS ? vs.z - c2 : 0.f;
    const float sd3 = hasS ? vs.w - c3 : 0.f;
    const float wd0 = (j0 > 0) ? vwesthalo - c0 : 0.f;
    const float wd1 = c0 - c1;
    const float wd2 = c1 - c2;
    const float wd3 = c2 - c3;
    const float ed0 = c1 - c0;
    const float ed1 = c2 - c1;
    const float ed2 = c3 - c2;
    const float ed3 = (j0 + 4 < W) ? veasthalo - c3 : 0.f;

    // ---- Drain async copies, read conduct back from this thread's own LDS
    // slot (no cross-thread sharing -> counter wait suffices, no barrier).
    asm volatile("s_wait_asynccnt 0x0" ::: "memory");
    const v4f c0a = *(const v4f*)(myslot + 0);
    const v4f c0b = *(const v4f*)(myslot + 4);
    const v4f c1a = *(const v4f*)(myslot + 8);
    const v4f c1b = *(const v4f*)(myslot + 12);

    // 2x2 conductance-weighted reduce per pixel.
    v4f cg;
    cg.x = (nd0 * c0a.x + wd0 * c0a.y) + (ed0 * c1a.x + sd0 * c1a.y);
    cg.y = (nd1 * c0a.z + wd1 * c0a.w) + (ed1 * c1a.z + sd1 * c1a.w);
    cg.z = (nd2 * c0b.x + wd2 * c0b.y) + (ed2 * c1b.x + sd2 * c1b.y);
    cg.w = (nd3 * c0b.z + wd3 * c0b.w) + (ed3 * c1b.z + sd3 * c1b.w);

    const v4f ca = {0.001f, 0.001f, 0.001f, 0.001f};   // point pixels fixed later
    const v4f cl = cg - ca;
    const v4f ls = cl * wgt;

    // Outputs are write-once -> non-temporal b128 stores (don't pollute L2).
    const size_t o = (size_t)i * W + (size_t)j0;
    __builtin_nontemporal_store(ls, (v4f*)(out + o));
    __builtin_nontemporal_store(cl, (v4f*)(out + (size_t)HW + o));
    __builtin_nontemporal_store(cg, (v4f*)(out + (size_t)2 * HW + o));
    __builtin_nontemporal_store(ca, (v4f*)(out + (size_t)3 * HW + o));
}

// At elvdd pad pixels: vmat_bound zeroes the pixel's own 2x2 vm block -> cg=0,
// and cur_active=0 -> cur_loss=0, loss=0. (Points are interior with >=2-cell
// margin, and the neighbor-direction writes are numerically identical to the
// stencil values, so no neighboring pixel changes.)
__global__ void point_fix_kernel(const int* __restrict__ point, int npts,
                                 float* __restrict__ out)
{
    const int p = blockIdx.x * blockDim.x + threadIdx.x;
    if (p >= npts) return;
    const int px = point[2 * p];
    const int py = point[2 * p + 1];
    const size_t o = (size_t)px * W + (size_t)py;
    out[o]                    = 0.f;   // loss
    out[(size_t)HW + o]       = 0.f;   // cur_loss
    out[(size_t)2 * HW + o]   = 0.f;   // cg
    out[(size_t)3 * HW + o]   = 0.f;   // cur_active
}

extern "C" void kernel_launch(void* const* d_in, const int* in_sizes, int n_in,
                              void* d_out, int out_size, void* d_ws, size_t ws_size,
                              hipStream_t stream)
{
    // setup_inputs order: v_data(0) [unused], v(1), conduct(2), weight(3),
    //                     type_mat(4) [forward no-op], point(5)
    const float* v       = (const float*)d_in[1];
    const float* conduct = (const float*)d_in[2];
    const float* weight  = (const float*)d_in[3];
    const int*   point   = (const int*)d_in[5];
    const int    npts    = in_sizes[5] / 2;

    float* out = (float*)d_out;

    dim3 block(256, 1, 1);
    dim3 grid(W / 4 / 256, H, 1);   // exact cover: 512 threads per row, EXEC all-1s
    grid_current_kernel<<<grid, block, 0, stream>>>(v, conduct, weight, out);

    point_fix_kernel<<<(npts + 255) / 256, 256, 0, stream>>>(point, npts, out);
}